// GAT_16174846836857
// MI455X (gfx1250) — compile-verified
//
#include <hip/hip_runtime.h>
#include <math.h>

#define NN    50000
#define EE    800000
#define ETOT  (EE + NN)      // edges + self loops
#define INC   128
#define CDIM  64             // per-head channels (both layers)
#define H1    4
#define F1    (H1 * CDIM)    // 256
#define F2    CDIM           // 64

typedef __attribute__((ext_vector_type(2))) float v2f;
typedef __attribute__((ext_vector_type(8))) float v8f;

static __device__ __forceinline__ int floatToKey(float f) {
  int i = __float_as_int(f);
  return (i >= 0) ? i : (i ^ 0x7fffffff);
}
static __device__ __forceinline__ float keyToFloat(int k) {
  int i = (k >= 0) ? k : (k ^ 0x7fffffff);
  return __int_as_float(i);
}

// ---------------------------------------------------------------------------
// C[M,Ncol] = A[M,K] * B[K,Ncol], fp32, one wave per 16x16 tile via
// V_WMMA_F32_16X16X4_F32.  A layout: M=lane%16, VGPR0/1 = K {0,1} (lanes 0-15)
// or K {2,3} (lanes 16-31).  B mirrored with N across lanes.  C: VGPR i holds
// rows i (lanes 0-15) and i+8 (lanes 16-31).
// ---------------------------------------------------------------------------
__global__ void wmma_gemm_f32(const float* __restrict__ A,
                              const float* __restrict__ B,
                              float* __restrict__ C, int M, int K, int Ncol) {
  int gtid  = blockIdx.x * blockDim.x + threadIdx.x;
  int wave  = gtid >> 5;                  // wave-uniform
  int lane  = threadIdx.x & 31;
  int ntile = Ncol >> 4;
  int tileM = wave / ntile;
  int tileN = wave - tileM * ntile;
  if (tileM * 16 >= M) return;            // uniform branch: skips WMMA entirely

  int half = lane >> 4;                   // 0: K pair {0,1}, 1: K pair {2,3}
  int l    = lane & 15;

  v8f acc = {};
  const float* Arow = A + (size_t)(tileM * 16 + l) * K + 2 * half;
  const float* Bcol = B + (size_t)(2 * half) * Ncol + tileN * 16 + l;

  for (int k0 = 0; k0 < K; k0 += 4) {
    v2f a, b;
    a.x = Arow[k0];
    a.y = Arow[k0 + 1];
    b.x = Bcol[(size_t)k0 * Ncol];
    b.y = Bcol[(size_t)(k0 + 1) * Ncol];
    acc = __builtin_amdgcn_wmma_f32_16x16x4_f32(false, a, false, b,
                                                (short)0, acc, false, false);
  }

  float* Crow = C + (size_t)(tileM * 16 + half * 8) * Ncol + tileN * 16 + l;
#pragma unroll
  for (int i = 0; i < 8; ++i) Crow[(size_t)i * Ncol] = acc[i];
}

// per-(node,head) attention logits: dot(xp[n,h,:], a_src[h,:]) etc.
__global__ void attn_logits(const float* __restrict__ xp,
                            const float* __restrict__ a_src,
                            const float* __restrict__ a_dst,
                            int H, float* __restrict__ als,
                            float* __restrict__ ald) {
  int t = blockIdx.x * blockDim.x + threadIdx.x;
  if (t >= NN * H) return;
  int n = t / H, h = t - n * H;
  const float* v  = xp + (size_t)n * H * CDIM + h * CDIM;
  const float* as = a_src + h * CDIM;
  const float* ad = a_dst + h * CDIM;
  float s = 0.f, d = 0.f;
#pragma unroll 8
  for (int c = 0; c < CDIM; ++c) { s += v[c] * as[c]; d += v[c] * ad[c]; }
  als[t] = s; ald[t] = d;
}

__global__ void fill_i32(int* p, long long n, int v) {
  long long t = blockIdx.x * (long long)blockDim.x + threadIdx.x;
  if (t < n) p[t] = v;
}
__global__ void fill_f32(float* p, long long n, float v) {
  long long t = blockIdx.x * (long long)blockDim.x + threadIdx.x;
  if (t < n) p[t] = v;
}
__global__ void bias_init(float* __restrict__ out, const float* __restrict__ b,
                          long long rows, int cols) {
  long long t = blockIdx.x * (long long)blockDim.x + threadIdx.x;
  if (t >= rows * cols) return;
  out[t] = b[(int)(t % cols)];
}

// pass A: per-edge score + LeakyReLU, segment max via monotone int key
__global__ void edge_score_max(const int* __restrict__ ei,
                               const float* __restrict__ als,
                               const float* __restrict__ ald,
                               float* __restrict__ esc,
                               int* __restrict__ mkey, int H) {
  int t = blockIdx.x * blockDim.x + threadIdx.x;
  if (t >= ETOT) return;
  int src, dst;
  if (t < EE) { src = ei[t]; dst = ei[EE + t]; } else { src = dst = t - EE; }
  for (int h = 0; h < H; ++h) {
    float v = als[src * H + h] + ald[dst * H + h];
    v = (v >= 0.f) ? v : 0.2f * v;              // LeakyReLU(0.2)
    esc[(size_t)t * H + h] = v;
    atomicMax(&mkey[dst * H + h], floatToKey(v));
  }
}

// pass B: exp(e - max), segment sum
__global__ void edge_exp_sum(const int* __restrict__ ei,
                             float* __restrict__ esc,
                             const int* __restrict__ mkey,
                             float* __restrict__ denom, int H) {
  int t = blockIdx.x * blockDim.x + threadIdx.x;
  if (t >= ETOT) return;
  int dst = (t < EE) ? ei[EE + t] : t - EE;
  for (int h = 0; h < H; ++h) {
    float mx = keyToFloat(mkey[dst * H + h]);
    float ex = expf(esc[(size_t)t * H + h] - mx);
    esc[(size_t)t * H + h] = ex;
    atomicAdd(&denom[dst * H + h], ex);
  }
}

// pass C: one wave per (edge, head); 64 channels -> 2 per lane, coalesced.
__global__ void edge_scatter(const int* __restrict__ ei,
                             const float* __restrict__ esc,
                             const float* __restrict__ denom,
                             const float* __restrict__ xp,
                             float* __restrict__ out, int H) {
  long long gtid = blockIdx.x * (long long)blockDim.x + threadIdx.x;
  long long wv   = gtid >> 5;
  int lane = threadIdx.x & 31;
  long long total = (long long)ETOT * H;
  if (wv >= total) return;
  int t = (int)(wv / H);
  int h = (int)(wv - (long long)t * H);
  int src, dst;
  if (t < EE) { src = ei[t]; dst = ei[EE + t]; } else { src = dst = t - EE; }
  float alpha = esc[(size_t)t * H + h] / denom[dst * H + h];
  const float* xs = xp  + (size_t)src * H * CDIM + h * CDIM;
  float*       od = out + (size_t)dst * H * CDIM + h * CDIM;
  atomicAdd(&od[lane],      alpha * xs[lane]);
  atomicAdd(&od[lane + 32], alpha * xs[lane + 32]);
}

__global__ void elu_ip(float* x, long long n) {
  long long t = blockIdx.x * (long long)blockDim.x + threadIdx.x;
  if (t >= n) return;
  float v = x[t];
  x[t] = (v > 0.f) ? v : (expf(v) - 1.f);
}

static inline unsigned cdiv(long long a, long long b) {
  return (unsigned)((a + b - 1) / b);
}

extern "C" void kernel_launch(void* const* d_in, const int* in_sizes, int n_in,
                              void* d_out, int out_size, void* d_ws, size_t ws_size,
                              hipStream_t stream) {
  const float* x   = (const float*)d_in[0];
  const int*   ei  = (const int*)  d_in[1];   // [2, E] int32
  const float* W1  = (const float*)d_in[2];
  const float* a1s = (const float*)d_in[3];
  const float* a1d = (const float*)d_in[4];
  const float* b1  = (const float*)d_in[5];
  const float* W2  = (const float*)d_in[6];
  const float* a2s = (const float*)d_in[7];
  const float* a2d = (const float*)d_in[8];
  const float* b2  = (const float*)d_in[9];
  float* out = (float*)d_out;

  // workspace layout
  char* p = (char*)d_ws;
  float* xp1  = (float*)p; p += (size_t)NN * F1 * 4;   // 51.2 MB
  float* h1   = (float*)p; p += (size_t)NN * F1 * 4;   // 51.2 MB (accum -> ELU -> h)
  float* xp2  = (float*)p; p += (size_t)NN * F2 * 4;   // 12.8 MB
  float* esc1 = (float*)p; p += (size_t)ETOT * H1 * 4; // 13.6 MB
  float* esc2 = (float*)p; p += (size_t)ETOT * 4;      //  3.4 MB
  float* al1s = (float*)p; p += (size_t)NN * H1 * 4;
  float* al1d = (float*)p; p += (size_t)NN * H1 * 4;
  float* den1 = (float*)p; p += (size_t)NN * H1 * 4;
  int*   mk1  = (int*)  p; p += (size_t)NN * H1 * 4;
  float* al2s = (float*)p; p += (size_t)NN * 4;
  float* al2d = (float*)p; p += (size_t)NN * 4;
  float* den2 = (float*)p; p += (size_t)NN * 4;
  int*   mk2  = (int*)  p; p += (size_t)NN * 4;

  const int B = 256;
  const int NEG_INF_KEY = (int)0x80000000;

  // ---- layer 1 ----
  {
    long long waves = (long long)(NN / 16) * (F1 / 16);            // 50000
    wmma_gemm_f32<<<cdiv(waves * 32, B), B, 0, stream>>>(x, W1, xp1, NN, INC, F1);
  }
  attn_logits<<<cdiv((long long)NN * H1, B), B, 0, stream>>>(xp1, a1s, a1d, H1, al1s, al1d);
  fill_i32<<<cdiv((long long)NN * H1, B), B, 0, stream>>>(mk1, (long long)NN * H1, NEG_INF_KEY);
  fill_f32<<<cdiv((long long)NN * H1, B), B, 0, stream>>>(den1, (long long)NN * H1, 0.f);
  bias_init<<<cdiv((long long)NN * F1, B), B, 0, stream>>>(h1, b1, NN, F1);

  edge_score_max<<<cdiv(ETOT, B), B, 0, stream>>>(ei, al1s, al1d, esc1, mk1, H1);
  edge_exp_sum  <<<cdiv(ETOT, B), B, 0, stream>>>(ei, esc1, mk1, den1, H1);
  edge_scatter  <<<cdiv((long long)ETOT * H1 * 32, B), B, 0, stream>>>(ei, esc1, den1, xp1, h1, H1);

  elu_ip<<<cdiv((long long)NN * F1, B), B, 0, stream>>>(h1, (long long)NN * F1);

  // ---- layer 2 ----
  {
    long long waves = (long long)(NN / 16) * (F2 / 16);            // 12500
    wmma_gemm_f32<<<cdiv(waves * 32, B), B, 0, stream>>>(h1, W2, xp2, NN, F1, F2);
  }
  attn_logits<<<cdiv((long long)NN, B), B, 0, stream>>>(xp2, a2s, a2d, 1, al2s, al2d);
  fill_i32<<<cdiv((long long)NN, B), B, 0, stream>>>(mk2, (long long)NN, NEG_INF_KEY);
  fill_f32<<<cdiv((long long)NN, B), B, 0, stream>>>(den2, (long long)NN, 0.f);
  bias_init<<<cdiv((long long)NN * F2, B), B, 0, stream>>>(out, b2, NN, F2);

  edge_score_max<<<cdiv(ETOT, B), B, 0, stream>>>(ei, al2s, al2d, esc2, mk2, 1);
  edge_exp_sum  <<<cdiv(ETOT, B), B, 0, stream>>>(ei, esc2, mk2, den2, 1);
  edge_scatter  <<<cdiv((long long)ETOT * 32, B), B, 0, stream>>>(ei, esc2, den2, xp2, out, 1);
}